// GCNModel_40750649704920
// MI455X (gfx1250) — compile-verified
//
#include <hip/hip_runtime.h>

typedef __attribute__((ext_vector_type(2))) float v2f;
typedef __attribute__((ext_vector_type(8))) float v8f;

#define N_NODES   50000
#define N_EDGES   800000
#define N_FEAT    128
#define HIDDEN    256
#define N_CLASSES 8
#define N_GRAPHS  64

// ---------------- workspace layout (bytes, 256B aligned) ----------------
// two ping-pong node-feature buffers (each 50000*256*4 = 51,200,000 B)
#define OFF_H    ((size_t)0)
#define OFF_X    ((size_t)51200000)
#define OFF_DIS  ((size_t)102400000)   // deg accumulation, then deg_inv_sqrt (200,000 B)
#define OFF_DINV ((size_t)102600192)   // deg_inv (200,000 B)
#define OFF_COEF ((size_t)102800384)   // per-edge norm coefficient (3,200,000 B)
#define OFF_CNT  ((size_t)106000384)   // per-graph node counts (256 B)
#define OFF_POOL ((size_t)106000640)   // pooled sums 64x256 (65,536 B)

// ---------------- setup kernels ----------------
__global__ void zero_kernel(float* __restrict__ dis, float* __restrict__ counts,
                            float* __restrict__ pooled) {
  int n = blockIdx.x * blockDim.x + threadIdx.x;
  if (n < N_NODES) dis[n] = 0.0f;
  if (n < N_GRAPHS * HIDDEN) pooled[n] = 0.0f;
  if (n < N_GRAPHS) counts[n] = 0.0f;
}

__global__ void deg_kernel(const int* __restrict__ edge, float* __restrict__ deg) {
  int e = blockIdx.x * blockDim.x + threadIdx.x;
  if (e < N_EDGES) atomicAdd(&deg[edge[N_EDGES + e]], 1.0f);
}

__global__ void deg_finish_kernel(float* __restrict__ dis, float* __restrict__ dinv) {
  int n = blockIdx.x * blockDim.x + threadIdx.x;
  if (n < N_NODES) {
    float d = dis[n] + 1.0f;       // self-loop included in degree
    dis[n]  = rsqrtf(d);
    dinv[n] = 1.0f / d;
  }
}

__global__ void coef_kernel(const int* __restrict__ edge, const float* __restrict__ dis,
                            float* __restrict__ coef) {
  int e = blockIdx.x * blockDim.x + threadIdx.x;
  if (e < N_EDGES) coef[e] = dis[edge[e]] * dis[edge[N_EDGES + e]];
}

__global__ void count_kernel(const int* __restrict__ batch, float* __restrict__ counts) {
  int n = blockIdx.x * blockDim.x + threadIdx.x;
  if (n < N_NODES) atomicAdd(&counts[batch[n]], 1.0f);
}

// ---------------- f32 WMMA GEMM: H[M,256] = act(X[M,K]) * W[K,256] ----------------
// One wave (32 threads) per 16x64 output tile: 4 independent accumulator chains
// (pipelines the XDL), 4x reuse of each A fragment. grid = (M/16, 256/64).
// A 16x4 layout (ISA 7.12.2): lanes 0-15 -> M=lane, VGPR0/1 = K+0,K+1;
//                             lanes 16-31 -> M=lane-16, VGPR0/1 = K+2,K+3.
// B 4x16 mirrored (N across lanes). C/D: VGPR r -> rows r (lanes 0-15) and r+8 (lanes 16-31).
template <int RELU>
__global__ void gemm_wmma_kernel(const float* __restrict__ X, const float* __restrict__ W,
                                 float* __restrict__ H, int K) {
  const int row0 = blockIdx.x << 4;
  const int col0 = blockIdx.y << 6;          // 64-wide column tile
  const int lane = threadIdx.x;
  const int l    = lane & 15;
  const int hi   = lane >> 4;                // 0 or 1
  const int koff = hi << 1;                  // 0 or 2
  const float* __restrict__ arow = X + (size_t)(row0 + l) * (size_t)K + koff;
  const float* __restrict__ bcol = W + (size_t)col0 + l;
  v8f c0 = {}, c1 = {}, c2 = {}, c3 = {};
  for (int k = 0; k < K; k += 4) {
    v2f a;
    a.x = arow[k];
    a.y = arow[k + 1];
    if (RELU) { a.x = fmaxf(a.x, 0.0f); a.y = fmaxf(a.y, 0.0f); }
    const float* __restrict__ bp0 = bcol + (size_t)(k + koff) * HIDDEN;
    const float* __restrict__ bp1 = bp0 + HIDDEN;
    v2f b0, b1, b2, b3;
    b0.x = bp0[0];   b0.y = bp1[0];
    b1.x = bp0[16];  b1.y = bp1[16];
    b2.x = bp0[32];  b2.y = bp1[32];
    b3.x = bp0[48];  b3.y = bp1[48];
    // (neg_a, A, neg_b, B, c_mod, C, reuse_a, reuse_b)
    c0 = __builtin_amdgcn_wmma_f32_16x16x4_f32(false, a, false, b0, (short)0, c0, false, false);
    c1 = __builtin_amdgcn_wmma_f32_16x16x4_f32(false, a, false, b1, (short)0, c1, false, false);
    c2 = __builtin_amdgcn_wmma_f32_16x16x4_f32(false, a, false, b2, (short)0, c2, false, false);
    c3 = __builtin_amdgcn_wmma_f32_16x16x4_f32(false, a, false, b3, (short)0, c3, false, false);
  }
  float* __restrict__ orow = H + (size_t)(row0 + (hi << 3)) * HIDDEN + col0 + l;
#pragma unroll
  for (int r = 0; r < 8; ++r) {
    orow[(size_t)r * HIDDEN +  0] = c0[r];
    orow[(size_t)r * HIDDEN + 16] = c1[r];
    orow[(size_t)r * HIDDEN + 32] = c2[r];
    orow[(size_t)r * HIDDEN + 48] = c3[r];
  }
}

// ---------------- OUT = H * deg_inv + b (self-loop term, seeds the scatter) ----------------
__global__ void init_kernel(const float* __restrict__ H, const float* __restrict__ dinv,
                            const float* __restrict__ b, float* __restrict__ OUT) {
  int idx = blockIdx.x * blockDim.x + threadIdx.x;   // N_NODES*256 = 12.8M threads
  if (idx < N_NODES * HIDDEN) {
    int n = idx >> 8;
    int cc = idx & (HIDDEN - 1);
    OUT[idx] = H[idx] * dinv[n] + b[cc];
  }
}

// ---------------- edge scatter: OUT[dst] += coef_e * H[src] ----------------
// 64 threads per edge, float4 gather (L2-resident), 4 no-return fp32 atomics each.
__global__ void scatter_kernel(const int* __restrict__ edge, const float* __restrict__ coef,
                               const float* __restrict__ H, float* __restrict__ OUT) {
  int tid = blockIdx.x * blockDim.x + threadIdx.x;   // N_EDGES*64 = 51.2M threads
  int e   = tid >> 6;
  if (e >= N_EDGES) return;
  int c4  = (tid & 63) << 2;                         // 0..252 step 4
  int s   = edge[e];
  int d   = edge[N_EDGES + e];
  float cf = coef[e];
  const float4 hv = *(const float4*)(H + (size_t)s * HIDDEN + c4);
  float* o = OUT + (size_t)d * HIDDEN + c4;
  atomicAdd(o + 0, hv.x * cf);
  atomicAdd(o + 1, hv.y * cf);
  atomicAdd(o + 2, hv.z * cf);
  atomicAdd(o + 3, hv.w * cf);
}

// ---------------- pooled[g] += relu(X[n]) ; batch is sorted so register-accumulate ----------
#define POOL_NB 8   // nodes per block (N_NODES % POOL_NB == 0)
__global__ void pool_kernel(const float* __restrict__ X, const int* __restrict__ batch,
                            float* __restrict__ pooled) {
  const int n0 = blockIdx.x * POOL_NB;
  const int c  = threadIdx.x;                        // 0..255
  float acc = 0.0f;
  int g = batch[n0];
#pragma unroll
  for (int i = 0; i < POOL_NB; ++i) {
    int n  = n0 + i;
    int gn = batch[n];
    if (gn != g) {                                   // graph boundary: flush partial sum
      atomicAdd(&pooled[g * HIDDEN + c], acc);
      acc = 0.0f;
      g = gn;
    }
    acc += fmaxf(X[(size_t)n * HIDDEN + c], 0.0f);
  }
  atomicAdd(&pooled[g * HIDDEN + c], acc);
}

// ---------------- out[64,8] = (pooled/counts) @ Wlin + blin ----------------
__global__ void final_kernel(const float* __restrict__ pooled, const float* __restrict__ counts,
                             const float* __restrict__ Wlin, const float* __restrict__ blin,
                             float* __restrict__ out) {
  int t = blockIdx.x * blockDim.x + threadIdx.x;
  if (t >= N_GRAPHS * N_CLASSES) return;
  int g = t >> 3, cls = t & 7;
  float inv = 1.0f / fmaxf(counts[g], 1.0f);
  float acc = blin[cls];
  for (int k = 0; k < HIDDEN; ++k)
    acc += pooled[g * HIDDEN + k] * inv * Wlin[k * N_CLASSES + cls];
  out[t] = acc;
}

// ---------------- host-side orchestration ----------------
extern "C" void kernel_launch(void* const* d_in, const int* in_sizes, int n_in,
                              void* d_out, int out_size, void* d_ws, size_t ws_size,
                              hipStream_t stream) {
  const float* x    = (const float*)d_in[0];
  const int*   edge = (const int*)d_in[1];
  const int*   batch= (const int*)d_in[2];
  const float* W1   = (const float*)d_in[3];
  const float* b1   = (const float*)d_in[4];
  const float* W2   = (const float*)d_in[5];
  const float* b2   = (const float*)d_in[6];
  const float* W3   = (const float*)d_in[7];
  const float* b3   = (const float*)d_in[8];
  const float* Wlin = (const float*)d_in[9];
  const float* blin = (const float*)d_in[10];
  float* out = (float*)d_out;

  char* ws = (char*)d_ws;
  float* H      = (float*)(ws + OFF_H);
  float* X      = (float*)(ws + OFF_X);
  float* dis    = (float*)(ws + OFF_DIS);
  float* dinv   = (float*)(ws + OFF_DINV);
  float* coef   = (float*)(ws + OFF_COEF);
  float* counts = (float*)(ws + OFF_CNT);
  float* pooled = (float*)(ws + OFF_POOL);

  const int TB = 256;
  const int nNodeBlk = (N_NODES + TB - 1) / TB;
  const int nEdgeBlk = (N_EDGES + TB - 1) / TB;
  const int nFeatBlk = (N_NODES * HIDDEN) / TB;      // 50,000 blocks
  const int nScatBlk = (N_EDGES * 64) / TB;          // 200,000 blocks
  const int nPoolBlk = N_NODES / POOL_NB;            // 6,250 blocks
  const dim3 gemmGrid(N_NODES / 16, HIDDEN / 64);    // 3125 x 4 waves

  // normalization precompute
  zero_kernel<<<nNodeBlk, TB, 0, stream>>>(dis, counts, pooled);
  deg_kernel<<<nEdgeBlk, TB, 0, stream>>>(edge, dis);
  deg_finish_kernel<<<nNodeBlk, TB, 0, stream>>>(dis, dinv);
  coef_kernel<<<nEdgeBlk, TB, 0, stream>>>(edge, dis, coef);
  count_kernel<<<nNodeBlk, TB, 0, stream>>>(batch, counts);

  // layer 1 (K=128, no input relu)
  gemm_wmma_kernel<0><<<gemmGrid, 32, 0, stream>>>(x, W1, H, N_FEAT);
  init_kernel<<<nFeatBlk, TB, 0, stream>>>(H, dinv, b1, X);
  scatter_kernel<<<nScatBlk, TB, 0, stream>>>(edge, coef, H, X);

  // layer 2 (K=256, relu folded into A loads)
  gemm_wmma_kernel<1><<<gemmGrid, 32, 0, stream>>>(X, W2, H, HIDDEN);
  init_kernel<<<nFeatBlk, TB, 0, stream>>>(H, dinv, b2, X);
  scatter_kernel<<<nScatBlk, TB, 0, stream>>>(edge, coef, H, X);

  // layer 3
  gemm_wmma_kernel<1><<<gemmGrid, 32, 0, stream>>>(X, W3, H, HIDDEN);
  init_kernel<<<nFeatBlk, TB, 0, stream>>>(H, dinv, b3, X);
  scatter_kernel<<<nScatBlk, TB, 0, stream>>>(edge, coef, H, X);

  // relu folded into pooling; then classifier
  pool_kernel<<<nPoolBlk, TB, 0, stream>>>(X, batch, pooled);
  final_kernel<<<2, TB, 0, stream>>>(pooled, counts, Wlin, blin, out);
}